// QCNNHybrid_65481071403339
// MI455X (gfx1250) — compile-verified
//
#include <hip/hip_runtime.h>
#include <hip/hip_bf16.h>

typedef float v2f __attribute__((ext_vector_type(2)));
typedef float v8f __attribute__((ext_vector_type(8)));

#define BATCH 8192
#define KSPLIT 8          // waves per block, each owns BATCH/KSPLIT keys
#define KEYS_PER_WAVE (BATCH / KSPLIT)   // 1024

// log2(e) * 0.5  (softmax scale 1/sqrt(4) folded with base-2 exp domain)
#define QSCALE 0.7213475204444817f

// ---------------------------------------------------------------------------
// Phase 1: per-sample tanh-MLP chain -> x[B,4], q = (0.5*log2e) * x@rot,
//          k = x@ent.  Scores then live in the log2 domain -> raw v_exp_f32.
// ---------------------------------------------------------------------------
__global__ __launch_bounds__(256) void qcnn_mlp_kernel(
    const float* __restrict__ in,                                   // [B,8]
    const float* __restrict__ W_fm, const float* __restrict__ b_fm, // 8x16
    const float* __restrict__ W_c1, const float* __restrict__ b_c1, // 16x16
    const float* __restrict__ W_p1, const float* __restrict__ b_p1, // 16x12
    const float* __restrict__ W_c2, const float* __restrict__ b_c2, // 12x8
    const float* __restrict__ W_p2, const float* __restrict__ b_p2, // 8x4
    const float* __restrict__ W_c3, const float* __restrict__ b_c3, // 4x4
    const float* __restrict__ rot,  const float* __restrict__ ent,  // 4x4
    float* __restrict__ xg, float* __restrict__ qg, float* __restrict__ kg)
{
  const int i = blockIdx.x * blockDim.x + threadIdx.x;
  if (i >= BATCH) return;

  float v_in[8];
  #pragma unroll
  for (int j = 0; j < 8; ++j) v_in[j] = in[i * 8 + j];

  float h0[16];
  #pragma unroll
  for (int o = 0; o < 16; ++o) {
    float s = b_fm[o];
    #pragma unroll
    for (int j = 0; j < 8; ++j) s += v_in[j] * W_fm[j * 16 + o];
    h0[o] = tanhf(s);
  }
  float h1[16];
  #pragma unroll
  for (int o = 0; o < 16; ++o) {
    float s = b_c1[o];
    #pragma unroll
    for (int j = 0; j < 16; ++j) s += h0[j] * W_c1[j * 16 + o];
    h1[o] = tanhf(s);
  }
  float h2[12];
  #pragma unroll
  for (int o = 0; o < 12; ++o) {
    float s = b_p1[o];
    #pragma unroll
    for (int j = 0; j < 16; ++j) s += h1[j] * W_p1[j * 12 + o];
    h2[o] = tanhf(s);
  }
  float h3[8];
  #pragma unroll
  for (int o = 0; o < 8; ++o) {
    float s = b_c2[o];
    #pragma unroll
    for (int j = 0; j < 12; ++j) s += h2[j] * W_c2[j * 8 + o];
    h3[o] = tanhf(s);
  }
  float h4[4];
  #pragma unroll
  for (int o = 0; o < 4; ++o) {
    float s = b_p2[o];
    #pragma unroll
    for (int j = 0; j < 8; ++j) s += h3[j] * W_p2[j * 4 + o];
    h4[o] = tanhf(s);
  }
  float x4[4];
  #pragma unroll
  for (int o = 0; o < 4; ++o) {
    float s = b_c3[o];
    #pragma unroll
    for (int j = 0; j < 4; ++j) s += h4[j] * W_c3[j * 4 + o];
    x4[o] = tanhf(s);
  }
  #pragma unroll
  for (int c = 0; c < 4; ++c) {
    float qv = 0.0f, kv = 0.0f;
    #pragma unroll
    for (int j = 0; j < 4; ++j) {
      qv += x4[j] * rot[j * 4 + c];
      kv += x4[j] * ent[j * 4 + c];
    }
    xg[i * 4 + c] = x4[c];
    qg[i * 4 + c] = QSCALE * qv;
    kg[i * 4 + c] = kv;
  }
}

// ---------------------------------------------------------------------------
// Phase 2: fused flash attention + conv + head.
// Block (256 thr = 8 waves) owns a 16-query tile. Each wave scans 1024 keys
// (64 tiles, unrolled x2). Scores via V_WMMA_F32_16X16X4_F32 computed
// TRANSPOSED: t = k_tile(16x4) x q^T(4x16) => D[key][query], so each lane
// owns one query and its 8 key-scores sit in registers -> the online-softmax
// max/sum are pure in-lane ops. Each 16-lane half keeps an independent
// (m, s, acc) state; halves and the 8 key-splits merge with exact
// flash-attention partial algebra (no per-tile cross-lane traffic).
// ---------------------------------------------------------------------------
__global__ __launch_bounds__(256) void qcnn_attn_kernel(
    const float* __restrict__ xg,   // [B,4]
    const float* __restrict__ qg,   // [B,4] pre-scaled (log2 domain)
    const float* __restrict__ kg,   // [B,4]
    const float* __restrict__ conv_w, const float* __restrict__ conv_b,
    const float* __restrict__ head_w, const float* __restrict__ head_b,
    float* __restrict__ out)        // [B]
{
  __shared__ float red[KSPLIT][16][6];   // per-split partials: m, s, acc[4]

  const int lane = threadIdx.x & 31;
  const int wi   = threadIdx.x >> 5;     // wave in block = key split id
  const int qb   = blockIdx.x * 16;      // 512 blocks x 16 queries
  const int half = lane >> 4;            // 0: keys r+0..7 ; 1: keys r+8..15
  const int l16  = lane & 15;            // this lane's query within the tile

  // B operand (q^T, 4x16): lane L<16 holds q[qb+L][0..1], lane>=16 holds
  // q[qb+L-16][2..3]  (16x4 f32 A/B VGPR layout, ISA 7.12.2).
  const v2f bq = *(const v2f*)(qg + (qb + l16) * 4 + 2 * half);

  float run_max = -3.0e38f;
  float run_sum = 0.0f;
  float acc0 = 0.f, acc1 = 0.f, acc2 = 0.f, acc3 = 0.f;

  const int kb0 = wi * KEYS_PER_WAVE;
  for (int kb = kb0; kb < kb0 + KEYS_PER_WAVE; kb += 32) {
    const v2f ak0 = *(const v2f*)(kg + (kb      + l16) * 4 + 2 * half);
    const v2f ak1 = *(const v2f*)(kg + (kb + 16 + l16) * 4 + 2 * half);
    v8f cz = {};
    v8f t0 = __builtin_amdgcn_wmma_f32_16x16x4_f32(
        false, ak0, false, bq, (short)0, cz, false, false);
    v8f t1 = __builtin_amdgcn_wmma_f32_16x16x4_f32(
        false, ak1, false, bq, (short)0, cz, false, false);

    // in-register max over this lane's 16 key-scores (both tiles)
    float m = t0[0];
    #pragma unroll
    for (int r = 1; r < 8; ++r) m = fmaxf(m, t0[r]);
    #pragma unroll
    for (int r = 0; r < 8; ++r) m = fmaxf(m, t1[r]);

    const float nm   = fmaxf(run_max, m);
    const float corr = __builtin_amdgcn_exp2f(run_max - nm);
    acc0 *= corr; acc1 *= corr; acc2 *= corr; acc3 *= corr;
    run_sum *= corr;
    run_max  = nm;

    const float* xr0 = xg + (kb      + half * 8) * 4;
    const float* xr1 = xg + (kb + 16 + half * 8) * 4;
    #pragma unroll
    for (int r = 0; r < 8; ++r) {
      const float p0 = __builtin_amdgcn_exp2f(t0[r] - nm);
      const float p1 = __builtin_amdgcn_exp2f(t1[r] - nm);
      const float4 xv0 = *(const float4*)(xr0 + r * 4);
      const float4 xv1 = *(const float4*)(xr1 + r * 4);
      run_sum += p0;
      run_sum += p1;
      acc0 += p0 * xv0.x; acc1 += p0 * xv0.y;
      acc2 += p0 * xv0.z; acc3 += p0 * xv0.w;
      acc0 += p1 * xv1.x; acc1 += p1 * xv1.y;
      acc2 += p1 * xv1.z; acc3 += p1 * xv1.w;
    }
  }

  // merge the two independent half states (same query) once per wave
  {
    const float mo = __shfl_xor(run_max, 16, 32);
    const float so = __shfl_xor(run_sum, 16, 32);
    const float a0 = __shfl_xor(acc0, 16, 32);
    const float a1 = __shfl_xor(acc1, 16, 32);
    const float a2 = __shfl_xor(acc2, 16, 32);
    const float a3 = __shfl_xor(acc3, 16, 32);
    const float M  = fmaxf(run_max, mo);
    const float c0 = __builtin_amdgcn_exp2f(run_max - M);
    const float c1 = __builtin_amdgcn_exp2f(mo - M);
    run_max = M;
    run_sum = run_sum * c0 + so * c1;
    acc0 = acc0 * c0 + a0 * c1;
    acc1 = acc1 * c0 + a1 * c1;
    acc2 = acc2 * c0 + a2 * c1;
    acc3 = acc3 * c0 + a3 * c1;
  }

  if (half == 0) {
    red[wi][l16][0] = run_max;
    red[wi][l16][1] = run_sum;
    red[wi][l16][2] = acc0;
    red[wi][l16][3] = acc1;
    red[wi][l16][4] = acc2;
    red[wi][l16][5] = acc3;
  }
  __syncthreads();

  // first 16 threads: merge the 8 key-split partials, finalize conv + head
  if (threadIdx.x < 16) {
    const int qi = threadIdx.x;
    float M = red[0][qi][0];
    #pragma unroll
    for (int s = 1; s < KSPLIT; ++s) M = fmaxf(M, red[s][qi][0]);
    float S = 0.f, A0 = 0.f, A1 = 0.f, A2 = 0.f, A3 = 0.f;
    #pragma unroll
    for (int s = 0; s < KSPLIT; ++s) {
      const float c = __builtin_amdgcn_exp2f(red[s][qi][0] - M);
      S  += red[s][qi][1] * c;
      A0 += red[s][qi][2] * c;
      A1 += red[s][qi][3] * c;
      A2 += red[s][qi][4] * c;
      A3 += red[s][qi][5] * c;
    }
    const float inv = 1.0f / S;
    const float o0 = A0 * inv, o1 = A1 * inv, o2 = A2 * inv, o3 = A3 * inv;
    // 2x2 conv (dot with conv_w) + bias, sigmoid
    float f = o0 * conv_w[0] + o1 * conv_w[1] + o2 * conv_w[2] + o3 * conv_w[3]
            + conv_b[0];
    f = 1.0f / (1.0f + __expf(-f));
    // head Linear(1,1) + sigmoid
    const float lg = f * head_w[0] + head_b[0];
    out[qb + qi] = 1.0f / (1.0f + __expf(-lg));
  }
}

// ---------------------------------------------------------------------------
extern "C" void kernel_launch(void* const* d_in, const int* in_sizes, int n_in,
                              void* d_out, int out_size, void* d_ws, size_t ws_size,
                              hipStream_t stream) {
  const float* in   = (const float*)d_in[0];
  const float* W_fm = (const float*)d_in[1];
  const float* b_fm = (const float*)d_in[2];
  const float* W_c1 = (const float*)d_in[3];
  const float* b_c1 = (const float*)d_in[4];
  const float* W_p1 = (const float*)d_in[5];
  const float* b_p1 = (const float*)d_in[6];
  const float* W_c2 = (const float*)d_in[7];
  const float* b_c2 = (const float*)d_in[8];
  const float* W_p2 = (const float*)d_in[9];
  const float* b_p2 = (const float*)d_in[10];
  const float* W_c3 = (const float*)d_in[11];
  const float* b_c3 = (const float*)d_in[12];
  const float* rot  = (const float*)d_in[13];
  const float* ent  = (const float*)d_in[14];
  const float* conv_w = (const float*)d_in[15];
  const float* conv_b = (const float*)d_in[16];
  const float* head_w = (const float*)d_in[17];
  const float* head_b = (const float*)d_in[18];

  float* ws = (float*)d_ws;
  float* xg = ws;                 // [B,4]
  float* qg = ws + BATCH * 4;     // [B,4]
  float* kg = ws + BATCH * 8;     // [B,4]

  qcnn_mlp_kernel<<<BATCH / 256, 256, 0, stream>>>(
      in, W_fm, b_fm, W_c1, b_c1, W_p1, b_p1, W_c2, b_c2,
      W_p2, b_p2, W_c3, b_c3, rot, ent, xg, qg, kg);

  // 512 blocks x 8 waves: each block = one 16-query tile, 8-way key split
  qcnn_attn_kernel<<<BATCH / 16, 256, 0, stream>>>(
      xg, qg, kg, conv_w, conv_b, head_w, head_b, (float*)d_out);
}